// VectorQuantizer_37873021616682
// MI455X (gfx1250) — compile-verified
//
#include <hip/hip_runtime.h>

typedef float v2f __attribute__((ext_vector_type(2)));
typedef float v8f __attribute__((ext_vector_type(8)));

#define B_   32
#define D_   64
#define H_   64
#define W_   64
#define K_   512
#define HW_  (H_*W_)          // 4096
#define CHW_ (D_*HW_)         // 262144
#define NPIX (B_*HW_)         // 131072
#define LDS_STRIDE 66         // padded codeword row stride (floats) -> bank-conflict-free B fetch
#define WAVES_PER_WG 8
#define PIX_PER_WAVE 16
#define NUM_WG 512
#define TOTAL_BLOCKS (NPIX / PIX_PER_WAVE)                  // 8192
#define BLOCKS_PER_WAVE (TOTAL_BLOCKS / (NUM_WG * WAVES_PER_WG))  // 2

// d_out layout: [quantized 8388608 f32][indices 131072 f32][loss 1 f32]
#define OUT_Q_ELEMS   (NPIX*D_)
#define OUT_IDX_ELEMS NPIX

__global__ void vq_zero_loss(float* loss) { *loss = 0.0f; }

__global__ __launch_bounds__(256)
void vq_kernel(const float* __restrict__ x,   // (B, D, H, W)
               const float* __restrict__ cw,  // (K, D)
               float* __restrict__ out_q,     // (B, D, H, W)
               float* __restrict__ out_idx,   // (B, H, W) as float
               float* __restrict__ out_loss)  // scalar
{
    extern __shared__ float smem[];
    float* lds_cw  = smem;                          // K_*LDS_STRIDE floats
    float* lds_csq = smem + K_*LDS_STRIDE;          // K_ floats

    const int tid  = threadIdx.x;
    const int lane = tid & 31;
    const int wave = tid >> 5;
    const int hf   = lane >> 4;   // wave half (K split for fp32 WMMA fragments)
    const int n    = lane & 15;   // pixel (A row) / codeword column (B col)

    // ---- stage all codewords into padded LDS + per-codeword squared norms ----
    #pragma unroll
    for (int j = 0; j < 2; ++j) {
        const int row = tid * 2 + j;                        // 512 rows / 256 threads
        const float4* src = (const float4*)(cw + row * D_);
        float2* dst = (float2*)(lds_cw + row * LDS_STRIDE); // 8B aligned (66 even)
        float s = 0.0f;
        #pragma unroll
        for (int q = 0; q < 16; ++q) {
            float4 v = src[q];
            s += v.x*v.x + v.y*v.y + v.z*v.z + v.w*v.w;
            dst[q*2+0] = make_float2(v.x, v.y);
            dst[q*2+1] = make_float2(v.z, v.w);
        }
        lds_csq[row] = s;
    }
    __syncthreads();   // only barrier in the kernel; waves independent below

    for (int it = 0; it < BLOCKS_PER_WAVE; ++it) {
        // ---- this wave's 16-pixel block (16 consecutive w -> contiguous) ----
        const int blk = (blockIdx.x * WAVES_PER_WG + wave) + it * (NUM_WG * WAVES_PER_WG);
        const int p0  = blk * PIX_PER_WAVE;
        const int b   = p0 >> 12;                           // / HW_
        const int hw0 = p0 & (HW_ - 1);
        const float* xbase = x + (size_t)b * CHW_ + hw0;

        // Preload A fragments (16x4 f32 per chunk, 16 chunks cover K=D=64).
        // Lanes 0-15 & 16-31 both hold M=0..15 (pixel = n);
        // VGPR0/1 carry K = {4c,4c+1} (half 0) or {4c+2,4c+3} (half 1).
        v2f afrag[16];
        #pragma unroll
        for (int c = 0; c < 16; ++c) {
            const int d = 4*c + 2*hf;
            afrag[c].x = xbase[(size_t)d       * HW_ + n];
            afrag[c].y = xbase[(size_t)(d + 1) * HW_ + n];
        }

        // Running argmin over score = ||c||^2 - 2*dot (||x||^2 const per pixel)
        float minv[8];
        int   mini[8];
        #pragma unroll
        for (int r = 0; r < 8; ++r) { minv[r] = 3.4e38f; mini[r] = 0; }

        for (int kt = 0; kt < K_ / 16; ++kt) {
            const int cid = kt * 16 + n;                    // this lane's codeword col
            const float* brow = lds_cw + cid * LDS_STRIDE + 2*hf;
            // two independent accumulators halve the WMMA RAW chain
            v8f acc0 = {0.f,0.f,0.f,0.f,0.f,0.f,0.f,0.f};
            v8f acc1 = {0.f,0.f,0.f,0.f,0.f,0.f,0.f,0.f};
            #pragma unroll
            for (int c = 0; c < 8; ++c) {
                v2f b0 = *(const v2f*)(brow + 8*c);         // ds_load_b64, bank-spread
                v2f b1 = *(const v2f*)(brow + 8*c + 4);
                acc0 = __builtin_amdgcn_wmma_f32_16x16x4_f32(
                           false, afrag[2*c],   false, b0, (short)0, acc0, false, false);
                acc1 = __builtin_amdgcn_wmma_f32_16x16x4_f32(
                           false, afrag[2*c+1], false, b1, (short)0, acc1, false, false);
            }
            const float csq = lds_csq[cid];
            #pragma unroll
            for (int r = 0; r < 8; ++r) {
                const float score = csq - 2.0f * (acc0[r] + acc1[r]); // row m = r+8*hf
                if (score < minv[r]) { minv[r] = score; mini[r] = cid; }
            }
        }

        // Cross-lane argmin within each 16-lane half (xor 1..8 stays in-half).
        #pragma unroll
        for (int r = 0; r < 8; ++r) {
            #pragma unroll
            for (int off = 1; off <= 8; off <<= 1) {
                const float ov = __shfl_xor(minv[r], off, 32);
                const int   oi = __shfl_xor(mini[r], off, 32);
                if (ov < minv[r] || (ov == minv[r] && oi < mini[r])) {
                    minv[r] = ov; mini[r] = oi;
                }
            }
        }
        // Every lane of half hf now holds mini[r] = argmin for row m = hf*8 + r.
        // Select this lane's pixel (n) index: cndmask tree on n&7, then swap halves.
        {
            const int r1 = n & 1, r2 = n & 2, r4 = n & 4;
            const int s0 = r1 ? mini[1] : mini[0];
            const int s1 = r1 ? mini[3] : mini[2];
            const int s2 = r1 ? mini[5] : mini[4];
            const int s3 = r1 ? mini[7] : mini[6];
            const int t0 = r2 ? s1 : s0;
            const int t1 = r2 ? s3 : s2;
            const int sel   = r4 ? t1 : t0;                 // row (hf*8 + (n&7))
            const int other = __shfl_xor(sel, 16, 32);      // same n, other half's rows
            const int myidx = ((n >> 3) == hf) ? sel : other;

            // ---- gather quantized vectors, write NCHW output, loss ----
            float* obase = out_q + (size_t)b * CHW_ + hw0;
            float lsum = 0.0f;
            #pragma unroll
            for (int i = 0; i < 32; ++i) {
                const int d = 2*i + hf;                     // lane: pixel n, 32 d's
                const float qv = lds_cw[myidx * LDS_STRIDE + d];
                const size_t off = (size_t)d * HW_ + n;
                const float xv = xbase[off];
                obase[off] = qv;                            // 64B contiguous / half-wave
                const float diff = qv - xv;
                lsum += diff * diff;
            }

            if (lane < 16)                                  // hf==0: myidx is pixel n's
                out_idx[p0 + lane] = (float)myidx;

            // loss = 1.25 * mean((q - x)^2) over N*D elements
            #pragma unroll
            for (int off = 16; off >= 1; off >>= 1)
                lsum += __shfl_xor(lsum, off, 32);
            if (lane == 0)
                atomicAdd(out_loss, lsum * (1.25f / (float)(NPIX * D_)));
        }
    }
}

extern "C" void kernel_launch(void* const* d_in, const int* in_sizes, int n_in,
                              void* d_out, int out_size, void* d_ws, size_t ws_size,
                              hipStream_t stream) {
    const float* x  = (const float*)d_in[0];   // (32, 64, 64, 64) f32
    const float* cw = (const float*)d_in[1];   // (512, 64) f32

    float* out      = (float*)d_out;
    float* out_q    = out;
    float* out_idx  = out + OUT_Q_ELEMS;
    float* out_loss = out + OUT_Q_ELEMS + OUT_IDX_ELEMS;

    vq_zero_loss<<<dim3(1), dim3(1), 0, stream>>>(out_loss);

    const size_t shmem = (size_t)(K_ * LDS_STRIDE + K_) * sizeof(float);
    vq_kernel<<<dim3(NUM_WG), dim3(256), shmem, stream>>>(x, cw, out_q, out_idx, out_loss);
}